// NHP_26568667693611
// MI455X (gfx1250) — compile-verified
//
#include <hip/hip_runtime.h>
#include <math.h>

typedef __attribute__((ext_vector_type(2))) float v2f;
typedef __attribute__((ext_vector_type(8))) float v8f;

#define N_NODES 20000
#define H_LINKS 2048
#define F_DIM   256
#define D_DIM   256
#define K_MEM   16
#define K_TOT   512   // fused K: [X | M] @ [W_loop ; W_aware]

// ---------------------------------------------------------------------------
// Kernel 1: zero the per-hyperlink member counters
// ---------------------------------------------------------------------------
__global__ void nhp_zero_counts(int* __restrict__ cnt) {
    int i = blockIdx.x * blockDim.x + threadIdx.x;
    if (i < H_LINKS) cnt[i] = 0;
}

// ---------------------------------------------------------------------------
// Kernel 2: recover member node indices from the incidence mask.
// mask is [N, H] row-major; exactly K ones per column. Coalesced scan with
// atomic slot assignment. Slot order is nondeterministic but max/min pooling
// is order-invariant, so the final output is deterministic.
// ---------------------------------------------------------------------------
__global__ void nhp_extract_members(const float* __restrict__ mask,
                                    int* __restrict__ cnt,
                                    int* __restrict__ idx) {
    long long i = (long long)blockIdx.x * blockDim.x + threadIdx.x;
    const long long total = (long long)N_NODES * H_LINKS;
    if (i >= total) return;
    if (mask[i] != 0.0f) {
        int h = (int)(i % H_LINKS);
        int n = (int)(i / H_LINKS);
        int p = atomicAdd(&cnt[h], 1);
        if (p < K_MEM) idx[h * K_MEM + p] = n;
    }
}

// ---------------------------------------------------------------------------
// Kernel 3: fused node-feature GEMM with fp32 WMMA.
//   H = relu(X @ W_loop + M @ W_aware + b_loop + b_aware)   [N, 256]
// Block = 256 threads = 8 wave32s. Block tile = 16 rows x 128 cols; each wave
// owns one 16x16 output tile (grid.y picks which 128-col half of D).
// A tile (16 rows x 64 K) staged in LDS, shared by all 8 waves.
// grid.z = 0 -> pos branch, 1 -> neg branch.
// ---------------------------------------------------------------------------
__global__ void __launch_bounds__(256, 2)
nhp_gemm_kernel(const float* __restrict__ Xp, const float* __restrict__ Mp,
                const float* __restrict__ Xn, const float* __restrict__ Mn,
                const float* __restrict__ Wl, const float* __restrict__ Wa,
                const float* __restrict__ bl, const float* __restrict__ ba,
                float* __restrict__ Hp, float* __restrict__ Hn) {
    // pad to 67: LDS bank for As[m][k] is (3m + k) mod 64 -> the WMMA A-frag
    // read pattern (lane = m, k offset 0/2 per half-wave) hits 32 distinct banks
    __shared__ float As[16][67];

    const float* X;
    const float* Mm;
    float* Hout;
    if (blockIdx.z == 0) { X = Xp; Mm = Mp; Hout = Hp; }
    else                 { X = Xn; Mm = Mn; Hout = Hn; }

    const int tid     = threadIdx.x;
    const int wave    = tid >> 5;        // 0..7
    const int lane    = tid & 31;
    const int rowbase = blockIdx.x * 16; // 20000/16 = 1250 exact tiles
    const int col     = blockIdx.y * 128 + wave * 16 + (lane & 15);
    const int mrow    = lane & 15;       // A fragment row (M index)
    const int khalf   = (lane >> 4) * 2; // K sub-offset per half-wave

    v8f acc = {};

    for (int chunk = 0; chunk < K_TOT / 64; ++chunk) {
        const int kbase = chunk * 64;
        // ---- cooperative A-tile stage: 16 rows x 64 K values (coalesced b128)
        {
            const float* src = (kbase < F_DIM) ? X : Mm;
            const int koff   = kbase & (F_DIM - 1);
            const int r      = tid >> 4;         // 0..15
            const int c4     = (tid & 15) * 4;   // 0..60
            const float4 v = *(const float4*)(src +
                (long long)(rowbase + r) * F_DIM + koff + c4);
            As[r][c4 + 0] = v.x;
            As[r][c4 + 1] = v.y;
            As[r][c4 + 2] = v.z;
            As[r][c4 + 3] = v.w;
        }
        __syncthreads();

        const float* W  = (kbase < F_DIM) ? Wl : Wa;
        const int    wk = kbase & (F_DIM - 1);

        #pragma unroll
        for (int kk = 0; kk < 64; kk += 4) {
            // A frag: 16x4 f32, lane=M, VGPR0/1 = K {khalf, khalf+1}
            v2f a;
            a.x = As[mrow][kk + khalf];
            a.y = As[mrow][kk + khalf + 1];
            // B frag: 4x16 f32, lane=N, VGPR0/1 = K {khalf, khalf+1}
            const float* wrow = W + (long long)(wk + kk + khalf) * D_DIM + col;
            v2f b;
            b.x = wrow[0];
            b.y = wrow[D_DIM];
            acc = __builtin_amdgcn_wmma_f32_16x16x4_f32(
                false, a, false, b, (short)0, acc, false, false);
        }
        __syncthreads();
    }

    // Epilogue: bias + relu + store. C/D layout: VGPR j holds M = j + 8*(lane>=16)
    const int rhalf = (lane >> 4) * 8;
    const float bias = bl[col] + ba[col];
    #pragma unroll
    for (int j = 0; j < 8; ++j) {
        float v = acc[j] + bias;
        v = v > 0.0f ? v : 0.0f;
        Hout[(long long)(rowbase + rhalf + j) * D_DIM + col] = v;
    }
}

// ---------------------------------------------------------------------------
// Kernel 4: max-min pooling over K members + score head.
// One block per (hyperlink, branch). Thread d handles hidden dim d.
// ---------------------------------------------------------------------------
__global__ void __launch_bounds__(256)
nhp_pool_score(const float* __restrict__ Hp, const float* __restrict__ Hn,
               const int* __restrict__ idx, const float* __restrict__ Ws,
               const float* __restrict__ bs, float* __restrict__ out) {
    const int h = blockIdx.x;
    const float* Hx = (blockIdx.y == 0) ? Hp : Hn;

    __shared__ int   members[K_MEM];
    __shared__ float red[8];

    const int tid  = threadIdx.x;
    const int lane = tid & 31;
    const int wave = tid >> 5;

    if (tid < K_MEM) members[tid] = idx[h * K_MEM + tid];
    __syncthreads();

    float mx = -3.402823466e38f;
    float mn =  3.402823466e38f;
    #pragma unroll
    for (int k = 0; k < K_MEM; ++k) {
        float v = Hx[(long long)members[k] * D_DIM + tid];
        mx = fmaxf(mx, v);
        mn = fminf(mn, v);
    }
    float partial = (mx - mn) * Ws[tid];

    // wave32 reduction
    #pragma unroll
    for (int off = 16; off > 0; off >>= 1)
        partial += __shfl_down(partial, off, 32);
    if (lane == 0) red[wave] = partial;
    __syncthreads();

    if (tid == 0) {
        float s = 0.0f;
        #pragma unroll
        for (int i = 0; i < 8; ++i) s += red[i];
        s += bs[0];
        out[blockIdx.y * H_LINKS + h] = 1.0f / (1.0f + expf(-s));
    }
}

// ---------------------------------------------------------------------------
extern "C" void kernel_launch(void* const* d_in, const int* in_sizes, int n_in,
                              void* d_out, int out_size, void* d_ws, size_t ws_size,
                              hipStream_t stream) {
    const float* pos_features = (const float*)d_in[0];
    const float* pos_matrix   = (const float*)d_in[1];
    const float* neg_features = (const float*)d_in[2];
    const float* neg_matrix   = (const float*)d_in[3];
    const float* batch_mask   = (const float*)d_in[4];
    const float* W_loop       = (const float*)d_in[5];
    const float* b_loop       = (const float*)d_in[6];
    const float* W_aware      = (const float*)d_in[7];
    const float* b_aware      = (const float*)d_in[8];
    const float* W_score      = (const float*)d_in[9];
    const float* b_score      = (const float*)d_in[10];

    // workspace layout
    float* Hp  = (float*)d_ws;                       // [N, D]
    float* Hn  = Hp + (size_t)N_NODES * D_DIM;       // [N, D]
    int*   idx = (int*)(Hn + (size_t)N_NODES * D_DIM); // [H, K]
    int*   cnt = idx + (size_t)H_LINKS * K_MEM;      // [H]

    float* out = (float*)d_out;  // [2048 pos | 2048 neg]

    // 1) reset counters
    nhp_zero_counts<<<(H_LINKS + 255) / 256, 256, 0, stream>>>(cnt);

    // 2) recover member indices from the mask (single coalesced 164 MB pass)
    {
        long long total = (long long)N_NODES * H_LINKS;
        int blocks = (int)((total + 255) / 256);
        nhp_extract_members<<<blocks, 256, 0, stream>>>(batch_mask, cnt, idx);
    }

    // 3) fused node-feature GEMMs (pos & neg) with fp32 WMMA
    {
        dim3 grid(N_NODES / 16, D_DIM / 128, 2);  // 1250 x 2 x 2
        nhp_gemm_kernel<<<grid, 256, 0, stream>>>(
            pos_features, pos_matrix, neg_features, neg_matrix,
            W_loop, W_aware, b_loop, b_aware, Hp, Hn);
    }

    // 4) max-min pool + sigmoid score head
    {
        dim3 grid(H_LINKS, 2);
        nhp_pool_score<<<grid, 256, 0, stream>>>(Hp, Hn, idx, W_score, b_score, out);
    }
}